// NonLocalAttentionBlock2D_75076028334858
// MI455X (gfx1250) — compile-verified
//
#include <hip/hip_runtime.h>
#include <math.h>

typedef __attribute__((ext_vector_type(2))) float v2f;
typedef __attribute__((ext_vector_type(8))) float v8f;

#define NB 4
#define CCH 64
#define ICH 32
#define NN 4096

__device__ __forceinline__ v8f wmma_f32(v2f a, v2f b, v8f c) {
  // D = A(16x4 f32) * B(4x16 f32) + C(16x16 f32)
  return __builtin_amdgcn_wmma_f32_16x16x4_f32(false, a, false, b, (short)0, c,
                                               false, false);
}

// ---------------- direct 3x3 stride-2 conv (tiny layers) ----------------
__global__ void conv3x3_s2_kernel(const float* __restrict__ in,
                                  const float* __restrict__ w,
                                  const float* __restrict__ bias,
                                  float* __restrict__ out,
                                  int Hin, int Win, int Hout, int Wout, int relu) {
  int idx = blockIdx.x * blockDim.x + threadIdx.x;
  int ow = idx % Wout;
  int t = idx / Wout;
  int oh = t % Hout; t /= Hout;
  int oc = t % CCH;
  int b = t / CCH;
  float acc = bias[oc];
  const float* wp = w + oc * CCH * 9;
  const float* ip = in + b * CCH * Hin * Win;
  for (int ic = 0; ic < CCH; ++ic) {
    const float* ipc = ip + ic * Hin * Win;
    const float* wpc = wp + ic * 9;
#pragma unroll
    for (int kh = 0; kh < 3; ++kh) {
      int ih = oh * 2 + kh - 1;
      if (ih < 0 || ih >= Hin) continue;
#pragma unroll
      for (int kw = 0; kw < 3; ++kw) {
        int iw = ow * 2 + kw - 1;
        if (iw < 0 || iw >= Win) continue;
        acc = fmaf(ipc[ih * Win + iw], wpc[kh * 3 + kw], acc);
      }
    }
  }
  if (relu) acc = (acc >= 0.f) ? acc : 0.2f * acc;
  out[idx] = acc;
}

// -------- bilinear x8 upsample (half-pixel, edge clamp) + sigmoid gate --------
__global__ void upsample_gate_kernel(const float* __restrict__ y8,
                                     const float* __restrict__ x,
                                     float* __restrict__ am) {
  int idx = blockIdx.x * blockDim.x + threadIdx.x;  // B*C*64*64
  int ow = idx & 63;
  int oh = (idx >> 6) & 63;
  int plane = idx >> 12;  // b*C + c
  float sy = (oh + 0.5f) * 0.125f - 0.5f;
  float sx = (ow + 0.5f) * 0.125f - 0.5f;
  int y0 = (int)floorf(sy);
  int x0 = (int)floorf(sx);
  float fy = sy - (float)y0, fx = sx - (float)x0;
  int y0c = min(max(y0, 0), 7), y1c = min(max(y0 + 1, 0), 7);
  int x0c = min(max(x0, 0), 7), x1c = min(max(x0 + 1, 0), 7);
  const float* p = y8 + plane * 64;
  float v00 = p[y0c * 8 + x0c], v01 = p[y0c * 8 + x1c];
  float v10 = p[y1c * 8 + x0c], v11 = p[y1c * 8 + x1c];
  float v0 = v00 + fx * (v01 - v00);
  float v1 = v10 + fx * (v11 - v10);
  float v = v0 + fy * (v1 - v0);
  float sig = 1.f / (1.f + expf(-v));
  am[idx] = sig * x[idx];
}

// ------------- fused 1x1 projections g/theta/phi via WMMA f32 -------------
// out_p[b, oc(32), n] = sum_c W_p[oc,c] * am[b,c,n] + bias_p[oc]
__global__ void proj_kernel(const float* __restrict__ am,
                            const float* __restrict__ gw, const float* __restrict__ gb,
                            const float* __restrict__ tw, const float* __restrict__ tb,
                            const float* __restrict__ pw, const float* __restrict__ pb,
                            float* __restrict__ go, float* __restrict__ to,
                            float* __restrict__ po) {
  const int lane = threadIdx.x & 31;
  const int tcol = lane & 15;
  const int half = lane >> 4;
  const int wid = (blockIdx.x * blockDim.x + threadIdx.x) >> 5;
  const int b = wid >> 8;
  const int n0 = (wid & 255) << 4;
  const float* Wm[3] = {gw, tw, pw};
  const float* Bs[3] = {gb, tb, pb};
  float* Om[3] = {go, to, po};
  v8f acc[3][2];
#pragma unroll
  for (int p = 0; p < 3; ++p)
#pragma unroll
    for (int mt = 0; mt < 2; ++mt)
#pragma unroll
      for (int r = 0; r < 8; ++r)
        acc[p][mt][r] = Bs[p][mt * 16 + r + 8 * half];
  const float* amb = am + b * CCH * NN;
#pragma unroll 4
  for (int ks = 0; ks < 16; ++ks) {
    int k0 = 4 * ks + 2 * half;
    v2f bf;
    bf.x = amb[k0 * NN + n0 + tcol];
    bf.y = amb[(k0 + 1) * NN + n0 + tcol];
#pragma unroll
    for (int p = 0; p < 3; ++p) {
#pragma unroll
      for (int mt = 0; mt < 2; ++mt) {
        v2f af;
        af.x = Wm[p][(mt * 16 + tcol) * CCH + k0];
        af.y = Wm[p][(mt * 16 + tcol) * CCH + k0 + 1];
        acc[p][mt] = wmma_f32(af, bf, acc[p][mt]);
      }
    }
  }
#pragma unroll
  for (int p = 0; p < 3; ++p)
#pragma unroll
    for (int mt = 0; mt < 2; ++mt)
#pragma unroll
      for (int r = 0; r < 8; ++r)
        Om[p][(b * ICH + mt * 16 + r + 8 * half) * NN + n0 + tcol] = acc[p][mt][r];
}

// ---- pass 1: per-column online softmax stats of f = theta^T phi (WMMA) ----
__global__ void colstats_kernel(const float* __restrict__ th,
                                const float* __restrict__ ph,
                                float* __restrict__ mcol, float* __restrict__ rden) {
  const int lane = threadIdx.x & 31;
  const int tcol = lane & 15;
  const int half = lane >> 4;
  const int wid = (blockIdx.x * blockDim.x + threadIdx.x) >> 5;
  const int b = wid >> 8;
  const int j0 = (wid & 255) << 4;
  const float* thb = th + b * ICH * NN;
  const float* phb = ph + b * ICH * NN;
  v2f bph[8];
#pragma unroll
  for (int ks = 0; ks < 8; ++ks) {
    int k0 = 4 * ks + 2 * half;
    bph[ks].x = phb[k0 * NN + j0 + tcol];
    bph[ks].y = phb[(k0 + 1) * NN + j0 + tcol];
  }
  float runM = -3.0e38f;
  float runS = 0.f;
  for (int it = 0; it < 256; ++it) {
    int i0 = it << 4;
    v8f F = {};
#pragma unroll
    for (int ks = 0; ks < 8; ++ks) {
      int k0 = 4 * ks + 2 * half;
      v2f a;
      a.x = thb[k0 * NN + i0 + tcol];
      a.y = thb[(k0 + 1) * NN + i0 + tcol];
      F = wmma_f32(a, bph[ks], F);
    }
    float tmax = F[0];
#pragma unroll
    for (int r = 1; r < 8; ++r) tmax = fmaxf(tmax, F[r]);
    float nm = fmaxf(runM, tmax);
    float s = 0.f;
#pragma unroll
    for (int r = 0; r < 8; ++r) s += expf(F[r] - nm);
    runS = runS * expf(runM - nm) + s;
    runM = nm;
  }
  // combine the two half-column partials (rows 0..7+16k live in lane, lane^16)
  float oM = __shfl_xor(runM, 16, 32);
  float oS = __shfl_xor(runS, 16, 32);
  float nm = fmaxf(runM, oM);
  float S = runS * expf(runM - nm) + oS * expf(oM - nm);
  if (half == 0) {
    mcol[b * NN + j0 + tcol] = nm;
    rden[b * NN + j0 + tcol] = 1.f / S;
  }
}

// ---- pass 2: y2[b,c,i] = sum_j exp(f[i,j]-m[j])*rden[j]*g[b,c,j] (WMMA) ----
__global__ void attn_kernel(const float* __restrict__ th, const float* __restrict__ ph,
                            const float* __restrict__ gg, const float* __restrict__ mcol,
                            const float* __restrict__ rden, float* __restrict__ y2) {
  __shared__ float lds[4 * 16 * 17];
  const int lane = threadIdx.x & 31;
  const int tcol = lane & 15;
  const int half = lane >> 4;
  const int wslot = (threadIdx.x >> 5) * (16 * 17);
  const int wid = (blockIdx.x * blockDim.x + threadIdx.x) >> 5;
  const int b = wid >> 8;
  const int i0 = (wid & 255) << 4;
  const float* thb = th + b * ICH * NN;
  const float* phb = ph + b * ICH * NN;
  const float* gb = gg + b * ICH * NN;
  v2f ath[8];  // theta^T A-fragments, reused across all j-tiles
#pragma unroll
  for (int ks = 0; ks < 8; ++ks) {
    int k0 = 4 * ks + 2 * half;
    ath[ks].x = thb[k0 * NN + i0 + tcol];
    ath[ks].y = thb[(k0 + 1) * NN + i0 + tcol];
  }
  v8f Y0 = {}, Y1 = {};
  for (int jt = 0; jt < 256; ++jt) {
    int j0 = jt << 4;
    v8f F = {};
#pragma unroll
    for (int ks = 0; ks < 8; ++ks) {
      int k0 = 4 * ks + 2 * half;
      v2f bp;
      bp.x = phb[k0 * NN + j0 + tcol];
      bp.y = phb[(k0 + 1) * NN + j0 + tcol];
      F = wmma_f32(ath[ks], bp, F);
    }
    if (jt + 1 < 256)
      __builtin_prefetch(&phb[(2 * half) * NN + j0 + 16 + tcol], 0, 3);
    float mj = mcol[b * NN + j0 + tcol];
    float rd = rden[b * NN + j0 + tcol];
    __syncthreads();
#pragma unroll
    for (int r = 0; r < 8; ++r)
      lds[wslot + (r + 8 * half) * 17 + tcol] = expf(F[r] - mj) * rd;
    __syncthreads();
#pragma unroll
    for (int st = 0; st < 4; ++st) {
      int k0 = 4 * st + 2 * half;
      v2f a;  // P tile re-read in A-fragment layout
      a.x = lds[wslot + tcol * 17 + k0];
      a.y = lds[wslot + tcol * 17 + k0 + 1];
      v2f bg0, bg1;  // g^T tiles (K=j, N=c)
      bg0.x = gb[tcol * NN + j0 + k0];
      bg0.y = gb[tcol * NN + j0 + k0 + 1];
      bg1.x = gb[(16 + tcol) * NN + j0 + k0];
      bg1.y = gb[(16 + tcol) * NN + j0 + k0 + 1];
      Y0 = wmma_f32(a, bg0, Y0);
      Y1 = wmma_f32(a, bg1, Y1);
    }
  }
#pragma unroll
  for (int r = 0; r < 8; ++r) {
    y2[(b * ICH + tcol) * NN + i0 + r + 8 * half] = Y0[r];
    y2[(b * ICH + 16 + tcol) * NN + i0 + r + 8 * half] = Y1[r];
  }
}

// ---- final: z = (W_w * y2 + w_b + am) * x  (1x1 conv GEMM via WMMA) ----
__global__ void final_kernel(const float* __restrict__ y2, const float* __restrict__ ww,
                             const float* __restrict__ wb, const float* __restrict__ am,
                             const float* __restrict__ x, float* __restrict__ out) {
  const int lane = threadIdx.x & 31;
  const int tcol = lane & 15;
  const int half = lane >> 4;
  const int wid = (blockIdx.x * blockDim.x + threadIdx.x) >> 5;
  const int b = wid >> 8;
  const int n0 = (wid & 255) << 4;
  const float* yb = y2 + b * ICH * NN;
  v8f acc[4];
#pragma unroll
  for (int mt = 0; mt < 4; ++mt)
#pragma unroll
    for (int r = 0; r < 8; ++r) acc[mt][r] = wb[mt * 16 + r + 8 * half];
#pragma unroll
  for (int ks = 0; ks < 8; ++ks) {
    int k0 = 4 * ks + 2 * half;
    v2f bf;
    bf.x = yb[k0 * NN + n0 + tcol];
    bf.y = yb[(k0 + 1) * NN + n0 + tcol];
#pragma unroll
    for (int mt = 0; mt < 4; ++mt) {
      v2f a;
      a.x = ww[(mt * 16 + tcol) * ICH + k0];
      a.y = ww[(mt * 16 + tcol) * ICH + k0 + 1];
      acc[mt] = wmma_f32(a, bf, acc[mt]);
    }
  }
#pragma unroll
  for (int mt = 0; mt < 4; ++mt) {
#pragma unroll
    for (int r = 0; r < 8; ++r) {
      int c = mt * 16 + r + 8 * half;
      int idx = (b * CCH + c) * NN + n0 + tcol;
      out[idx] = (acc[mt][r] + am[idx]) * x[idx];
    }
  }
}

extern "C" void kernel_launch(void* const* d_in, const int* in_sizes, int n_in,
                              void* d_out, int out_size, void* d_ws, size_t ws_size,
                              hipStream_t stream) {
  (void)in_sizes; (void)n_in; (void)out_size; (void)ws_size;
  const float* x   = (const float*)d_in[0];
  const float* d1w = (const float*)d_in[1];
  const float* d1b = (const float*)d_in[2];
  const float* d2w = (const float*)d_in[3];
  const float* d2b = (const float*)d_in[4];
  const float* d3w = (const float*)d_in[5];
  const float* d3b = (const float*)d_in[6];
  const float* gw  = (const float*)d_in[7];
  const float* gbv = (const float*)d_in[8];
  const float* tw  = (const float*)d_in[9];
  const float* tbv = (const float*)d_in[10];
  const float* pw  = (const float*)d_in[11];
  const float* pbv = (const float*)d_in[12];
  const float* ww  = (const float*)d_in[13];
  const float* wbv = (const float*)d_in[14];

  float* ws = (float*)d_ws;
  float* t1 = ws;                       // [4,64,32,32] = 262144
  float* t2 = t1 + 4 * 64 * 32 * 32;    // [4,64,16,16] = 65536
  float* t3 = t2 + 4 * 64 * 16 * 16;    // [4,64,8,8]   = 16384
  float* am = t3 + 4 * 64 * 8 * 8;      // [4,64,4096]  = 1048576
  float* gg = am + 4 * 64 * 4096;       // [4,32,4096]  = 524288
  float* th = gg + 4 * 32 * 4096;       // [4,32,4096]
  float* ph = th + 4 * 32 * 4096;       // [4,32,4096]
  float* mc = ph + 4 * 32 * 4096;       // [4,4096]
  float* rd = mc + 4 * 4096;            // [4,4096]
  float* y2 = rd + 4 * 4096;            // [4,32,4096]

  conv3x3_s2_kernel<<<1024, 256, 0, stream>>>(x, d1w, d1b, t1, 64, 64, 32, 32, 1);
  conv3x3_s2_kernel<<<256, 256, 0, stream>>>(t1, d2w, d2b, t2, 32, 32, 16, 16, 1);
  conv3x3_s2_kernel<<<64, 256, 0, stream>>>(t2, d3w, d3b, t3, 16, 16, 8, 8, 0);
  upsample_gate_kernel<<<4096, 256, 0, stream>>>(t3, x, am);
  proj_kernel<<<256, 128, 0, stream>>>(am, gw, gbv, tw, tbv, pw, pbv, gg, th, ph);
  colstats_kernel<<<256, 128, 0, stream>>>(th, ph, mc, rd);
  attn_kernel<<<256, 128, 0, stream>>>(th, ph, gg, mc, rd, y2);
  final_kernel<<<256, 128, 0, stream>>>(y2, ww, wbv, am, x, (float*)d_out);
}